// CostVolume_3393024164441
// MI455X (gfx1250) — compile-verified
//
#include <hip/hip_runtime.h>
#include <hip/hip_bf16.h>
#include <stdint.h>

// Problem constants
#define GTOT 40
#define CG   8
#define HIMG 128
#define WIMG 240
#define DTOT 48

// Tile geometry
#define TH 8            // output rows per workgroup
#define TW 40           // output cols per workgroup (240 = 6*40)
#define CRH 16          // corr rows   = TH+8 (halo 4 each side)
#define CRW 48          // corr cols   = TW+8
#define RW  96          // right cols  = CRW + 48 (disparity span), u0 = w0-52
#define CORR_PITCH 772  // 16*48 + 4 pad (spreads scatter across banks)
#define PATCH_H 14      // TH+6
#define PATCH_W 46      // TW+6
#define PATCH_PITCH 48

typedef float v2f __attribute__((ext_vector_type(2)));
typedef float v4f __attribute__((ext_vector_type(4)));
typedef float v8f __attribute__((ext_vector_type(8)));
// Matches the builtin's declared parameter pointee: gcc-style 4 x i32 vector
typedef int v4i_b __attribute__((vector_size(16)));

#if defined(__gfx1250__) && __has_builtin(__builtin_amdgcn_global_load_async_to_lds_b128)
#define ASYNC_OK 1
#define AS1 __attribute__((address_space(1)))
#define AS3 __attribute__((address_space(3)))
#else
#define ASYNC_OK 0
#endif

__global__ __launch_bounds__(256, 1)
void fused_gwc_patch_kernel(const float* __restrict__ left,
                            const float* __restrict__ right,
                            const float* __restrict__ w_patch,
                            const float* __restrict__ w_l1,
                            const float* __restrict__ w_l2,
                            const float* __restrict__ w_l3,
                            float* __restrict__ out)
{
    __shared__ float Ls[CG * CRH * CRW];                    //  24576 B
    __shared__ float Rs[CG * CRH * RW];                     //  49152 B
    __shared__ float Corr[15 * CORR_PITCH + CRH * CRW];     //  49392 B
    __shared__ float Patch[PATCH_H * PATCH_PITCH];          //   2688 B

    const int w0 = blockIdx.x * TW;
    const int h0 = blockIdx.y * TH;
    const int g  = blockIdx.z;
    const int tid  = threadIdx.x;
    const int lane = tid & 31;
    const int wv   = tid >> 5;

    // Per-group weights: wp = w_patch[g], w2/dil = second-level dwconv
    float wp[9], w2[9];
    int dil;
    {
        const float* p = w_patch + g * 9;
#pragma unroll
        for (int i = 0; i < 9; ++i) wp[i] = p[i];
        const float* q;
        if (g < 8)       { dil = 1; q = w_l1 + g * 9; }
        else if (g < 24) { dil = 2; q = w_l2 + (g - 8) * 9; }
        else             { dil = 3; q = w_l3 + (g - 24) * 9; }
#pragma unroll
        for (int i = 0; i < 9; ++i) w2[i] = q[i];
    }

    const float* Lbase = left  + (size_t)g * CG * HIMG * WIMG;
    const float* Rbase = right + (size_t)g * CG * HIMG * WIMG;
    const int u0 = w0 - 52;

    // ---- Stage L/R tiles into LDS (zero-padded outside image) ----
#if ASYNC_OK
    {
        v4f z = {0.f, 0.f, 0.f, 0.f};
        for (int i = tid; i < (CG*CRH*CRW)/4; i += 256) ((v4f*)Ls)[i] = z;
        for (int i = tid; i < (CG*CRH*RW)/4;  i += 256) ((v4f*)Rs)[i] = z;
        __syncthreads();   // zero-fill visible before async DMA overwrites
        // Left: 8ch x 16r x 48c, b128 per lane; 4-wide blocks are fully in/out
        for (int i = tid; i < CG*CRH*(CRW/4); i += 256) {
            int ww = (i % (CRW/4)) * 4;
            int hh = (i / (CRW/4)) % CRH;
            int c  =  i / ((CRW/4) * CRH);
            int h = h0 - 4 + hh;
            int w = w0 - 4 + ww;
            if ((unsigned)h < HIMG && (unsigned)w < WIMG) {
                const float* src = Lbase + ((size_t)c * HIMG + h) * WIMG + w;
                __builtin_amdgcn_global_load_async_to_lds_b128(
                    (AS1 v4i_b*)src,
                    (AS3 v4i_b*)&Ls[(c * CRH + hh) * CRW + ww],
                    0, 0);
            }
        }
        // Right: 8ch x 16r x 96c (u = w0-52 .. w0+43)
        for (int i = tid; i < CG*CRH*(RW/4); i += 256) {
            int uu = (i % (RW/4)) * 4;
            int hh = (i / (RW/4)) % CRH;
            int c  =  i / ((RW/4) * CRH);
            int h = h0 - 4 + hh;
            int u = u0 + uu;
            if ((unsigned)h < HIMG && (unsigned)u < WIMG) {
                const float* src = Rbase + ((size_t)c * HIMG + h) * WIMG + u;
                __builtin_amdgcn_global_load_async_to_lds_b128(
                    (AS1 v4i_b*)src,
                    (AS3 v4i_b*)&Rs[(c * CRH + hh) * RW + uu],
                    0, 0);
            }
        }
#if __has_builtin(__builtin_amdgcn_s_wait_asynccnt)
        __builtin_amdgcn_s_wait_asynccnt(0);
#else
        asm volatile("s_wait_asynccnt 0" ::: "memory");
#endif
    }
#else
    {
        v4f z = {0.f, 0.f, 0.f, 0.f};
        for (int i = tid; i < CG*CRH*(CRW/4); i += 256) {
            int ww = (i % (CRW/4)) * 4;
            int hh = (i / (CRW/4)) % CRH;
            int c  =  i / ((CRW/4) * CRH);
            int h = h0 - 4 + hh;
            int w = w0 - 4 + ww;
            v4f v = z;
            if ((unsigned)h < HIMG && (unsigned)w < WIMG)
                v = *(const v4f*)(Lbase + ((size_t)c * HIMG + h) * WIMG + w);
            *(v4f*)&Ls[(c * CRH + hh) * CRW + ww] = v;
        }
        for (int i = tid; i < CG*CRH*(RW/4); i += 256) {
            int uu = (i % (RW/4)) * 4;
            int hh = (i / (RW/4)) % CRH;
            int c  =  i / ((RW/4) * CRH);
            int h = h0 - 4 + hh;
            int u = u0 + uu;
            v4f v = z;
            if ((unsigned)h < HIMG && (unsigned)u < WIMG)
                v = *(const v4f*)(Rbase + ((size_t)c * HIMG + h) * WIMG + u);
            *(v4f*)&Rs[(c * CRH + hh) * RW + uu] = v;
        }
    }
#endif
    __syncthreads();

    // A/B operand lane mapping for V_WMMA_F32_16X16X4_F32:
    //   VGPR0: lanes 0-15 -> K=0, lanes 16-31 -> K=2 ; VGPR1: K=1 / K=3
    const int m  = lane & 15;          // also the B-matrix column index n
    const int k0 = (lane >> 4) * 2;

    for (int chunk = 0; chunk < 3; ++chunk) {
        const int d0 = chunk * 16;

        // ---- Correlation as band-matmul: S = L^T R (K = Cg = 8) ----
        // 48 (row, w-subtile) pairs, 6 per wave; two u-tiles per pair cover
        // disparities d0..d0+15 exactly (tile B: M>=N, tile A: M<N).
        for (int pi = 0; pi < 6; ++pi) {
            const int p   = wv + 8 * pi;       // 0..47
            const int hh  = p & 15;            // corr row
            const int wt  = (p >> 4) * 16;     // 0,16,32 within CRW=48
            const int uuB = 48 + wt - d0;      // Rs col of u-tile B (>=0)
            const int uuA = uuB - 16;

            v2f a0, a1, bA0, bA1, bB0, bB1;
            a0[0]  = Ls[((k0 + 0) * CRH + hh) * CRW + wt + m];
            a0[1]  = Ls[((k0 + 1) * CRH + hh) * CRW + wt + m];
            a1[0]  = Ls[((k0 + 4) * CRH + hh) * CRW + wt + m];
            a1[1]  = Ls[((k0 + 5) * CRH + hh) * CRW + wt + m];
            bA0[0] = Rs[((k0 + 0) * CRH + hh) * RW + uuA + m];
            bA0[1] = Rs[((k0 + 1) * CRH + hh) * RW + uuA + m];
            bA1[0] = Rs[((k0 + 4) * CRH + hh) * RW + uuA + m];
            bA1[1] = Rs[((k0 + 5) * CRH + hh) * RW + uuA + m];
            bB0[0] = Rs[((k0 + 0) * CRH + hh) * RW + uuB + m];
            bB0[1] = Rs[((k0 + 1) * CRH + hh) * RW + uuB + m];
            bB1[0] = Rs[((k0 + 4) * CRH + hh) * RW + uuB + m];
            bB1[1] = Rs[((k0 + 5) * CRH + hh) * RW + uuB + m];

            v8f accA = {0.f,0.f,0.f,0.f,0.f,0.f,0.f,0.f};
            v8f accB = {0.f,0.f,0.f,0.f,0.f,0.f,0.f,0.f};
            accA = __builtin_amdgcn_wmma_f32_16x16x4_f32(false, a0, false, bA0,
                                                         (short)0, accA, false, false);
            accA = __builtin_amdgcn_wmma_f32_16x16x4_f32(false, a1, false, bA1,
                                                         (short)0, accA, false, false);
            accB = __builtin_amdgcn_wmma_f32_16x16x4_f32(false, a0, false, bB0,
                                                         (short)0, accB, false, false);
            accB = __builtin_amdgcn_wmma_f32_16x16x4_f32(false, a1, false, bB1,
                                                         (short)0, accB, false, false);

            // Scatter anti-diagonal band: each lane owns 8 elements (one per
            // acc VGPR): M = v + 8*(lane>=16), N = lane&15, d = (M-N)&15 + d0.
            const int mhi = (lane >> 4) * 8;
#pragma unroll
            for (int v = 0; v < 8; ++v) {
                const int M  = v + mhi;
                const int dl = (M - m) & 15;
                const float val = (M >= m) ? accB[v] : accA[v];
                Corr[dl * CORR_PITCH + hh * CRW + wt + M] = val * 0.125f;  // mean over Cg=8
            }
        }
        __syncthreads();

        // ---- Per-disparity 2D conv chain (kernels have extent 1 in D) ----
        for (int dl = 0; dl < 16; ++dl) {
            const int d = d0 + dl;
            const float* cp = &Corr[dl * CORR_PITCH];
            // conv1: w_patch 3x3 dil1 on 16x48 corr -> 14x46 patch
            for (int idx = tid; idx < PATCH_H * PATCH_W; idx += 256) {
                const int ph = idx / PATCH_W;
                const int pw = idx - ph * PATCH_W;
                float s = 0.f;
#pragma unroll
                for (int i = 0; i < 3; ++i)
#pragma unroll
                    for (int j = 0; j < 3; ++j)
                        s = fmaf(wp[i*3+j], cp[(ph + i) * CRW + (pw + j)], s);
                Patch[ph * PATCH_PITCH + pw] = s;
            }
            __syncthreads();
            // conv2: 3x3 with dilation dil on 14x46 patch -> 8x40 output
            for (int idx = tid; idx < TH * TW; idx += 256) {
                const int oh = idx / TW;
                const int ow = idx - oh * TW;
                float s = 0.f;
#pragma unroll
                for (int i = 0; i < 3; ++i)
#pragma unroll
                    for (int j = 0; j < 3; ++j)
                        s = fmaf(w2[i*3+j],
                                 Patch[(oh + 3 + (i-1)*dil) * PATCH_PITCH +
                                       (ow + 3 + (j-1)*dil)], s);
                out[(((size_t)g * DTOT + d) * HIMG + (h0 + oh)) * WIMG + (w0 + ow)] = s;
            }
            __syncthreads();   // protects Patch (next dl) and Corr (next chunk)
        }
    }
}

extern "C" void kernel_launch(void* const* d_in, const int* in_sizes, int n_in,
                              void* d_out, int out_size, void* d_ws, size_t ws_size,
                              hipStream_t stream) {
    const float* left    = (const float*)d_in[0];   // left_gwc  [1,320,128,240]
    const float* right   = (const float*)d_in[1];   // right_gwc [1,320,128,240]
    // d_in[2], d_in[3] (left_unary/right_unary) are unused by the reference
    const float* w_patch = (const float*)d_in[4];   // [40,1,1,3,3]
    const float* w_l1    = (const float*)d_in[5];   // [8,1,1,3,3]
    const float* w_l2    = (const float*)d_in[6];   // [16,1,1,3,3]
    const float* w_l3    = (const float*)d_in[7];   // [16,1,1,3,3]
    float* out = (float*)d_out;                     // [1,40,48,128,240]

    dim3 grid(WIMG / TW, HIMG / TH, GTOT);          // 6 x 16 x 40
    fused_gwc_patch_kernel<<<grid, 256, 0, stream>>>(left, right, w_patch,
                                                     w_l1, w_l2, w_l3, out);
    (void)in_sizes; (void)n_in; (void)out_size; (void)d_ws; (void)ws_size;
}